// Hierarchical_classifer_new_42331197669650
// MI455X (gfx1250) — compile-verified
//
#include <hip/hip_runtime.h>

typedef float v2f __attribute__((ext_vector_type(2)));
typedef float v8f __attribute__((ext_vector_type(8)));

#define N_NODES 20000
#define L_LEV   10
#define S_LEV   2000
#define P_PAR   4
#define B_BATCH 64
#define D_IN    1024
#define H_DIM   128
#define N_EDGE  ((N_NODES - S_LEV) * P_PAR)   // 72000

// ---------------------------------------------------------------------------
// Kernel 1: s_in[b] = sum_d inputs[b,d], via V_WMMA_F32_16X16X4_F32.
// One wave per 16 batch rows. A = 16x4 f32 input tile, B = 4x16 ones.
// D[m,n] accumulates the row sum of m in every column n (exact f32 MACs).
// ---------------------------------------------------------------------------
__global__ __launch_bounds__(32) void rowsum_wmma_kernel(const float* __restrict__ in,
                                                         float* __restrict__ s_in)
{
    const int lane = threadIdx.x;        // 0..31, wave32
    const int m    = lane & 15;          // A-matrix row (M)
    const int half = lane >> 4;          // K-half per ISA 16x4 layout
    const int row0 = blockIdx.x << 4;

    v8f acc = {0.f, 0.f, 0.f, 0.f, 0.f, 0.f, 0.f, 0.f};
    v2f ones; ones[0] = 1.0f; ones[1] = 1.0f;

    const float* rowp = in + (size_t)(row0 + m) * D_IN + (half << 1);
    for (int k = 0; k < D_IN; k += 4) {
        v2f a;
        a[0] = rowp[k];
        a[1] = rowp[k + 1];
        // 8 args: (neg_a, A, neg_b, B, c_mod, C, reuse_a, reuse_b)
        acc = __builtin_amdgcn_wmma_f32_16x16x4_f32(false, a, false, ones,
                                                    (short)0, acc, false, false);
    }
    // D layout: VGPR r, lanes 0-15 -> M=r; lanes 16-31 -> M=8+r. Column N=0 suffices.
    if (m == 0) {
        const int mb = row0 + (half << 3);
        #pragma unroll
        for (int r = 0; r < 8; ++r) s_in[mb + r] = acc[r];
    }
}

// ---------------------------------------------------------------------------
// Kernel 2: per-node piecewise-linear coefficients from w1, w2.
//   apos = sum max(w1,0)        aneg = sum min(w1,0)
//   cpos = sum max(w1,0)*w2     cneg = sum min(w1,0)*w2
// One wave per node; each lane owns one float4 (H=128 = 32 lanes * 4).
// ---------------------------------------------------------------------------
__global__ __launch_bounds__(256) void coeff_kernel(const float* __restrict__ w1,
                                                    const float* __restrict__ w2,
                                                    float* __restrict__ apos,
                                                    float* __restrict__ aneg,
                                                    float* __restrict__ cpos,
                                                    float* __restrict__ cneg)
{
    const int gtid = blockIdx.x * blockDim.x + threadIdx.x;
    const int node = gtid >> 5;
    const int lane = threadIdx.x & 31;
    if (node >= N_NODES) return;

    const float4 a = ((const float4*)(w1 + (size_t)node * H_DIM))[lane];
    const float4 c = ((const float4*)(w2 + (size_t)node * H_DIM))[lane];

    const float px = fmaxf(a.x, 0.f), py = fmaxf(a.y, 0.f),
                pz = fmaxf(a.z, 0.f), pw = fmaxf(a.w, 0.f);
    const float nx = fminf(a.x, 0.f), ny = fminf(a.y, 0.f),
                nz = fminf(a.z, 0.f), nw = fminf(a.w, 0.f);

    float ap = px + py + pz + pw;
    float an = nx + ny + nz + nw;
    float cp = px * c.x + py * c.y + pz * c.z + pw * c.w;
    float cn = nx * c.x + ny * c.y + nz * c.z + nw * c.w;

    #pragma unroll
    for (int off = 16; off > 0; off >>= 1) {
        ap += __shfl_down(ap, off);
        an += __shfl_down(an, off);
        cp += __shfl_down(cp, off);
        cn += __shfl_down(cn, off);
    }
    if (lane == 0) {
        apos[node] = ap; aneg[node] = an;
        cpos[node] = cp; cneg[node] = cn;
    }
}

// ---------------------------------------------------------------------------
// Kernels 3..12: one topological level. Layout [node, b] so the 64 b-lanes
// of a node are coalesced for every parent gather.
// ---------------------------------------------------------------------------
__global__ __launch_bounds__(256) void level_kernel(const float* __restrict__ s_in,
                                                    const float* __restrict__ apos,
                                                    const float* __restrict__ aneg,
                                                    const float* __restrict__ cpos,
                                                    const float* __restrict__ cneg,
                                                    const float* __restrict__ b2,
                                                    const int*   __restrict__ parents,
                                                    float* __restrict__ net_sum,
                                                    float* __restrict__ outv,
                                                    float* __restrict__ fin,
                                                    int lo)
{
    const int idx = blockIdx.x * blockDim.x + threadIdx.x;   // S_LEV*B = 128000 exact
    const int b   = idx & (B_BATCH - 1);
    const int n   = lo + (idx >> 6);

    float s = s_in[b];
    if (lo > 0) {
        const int* par = parents + (size_t)n * P_PAR;
        #pragma unroll
        for (int p = 0; p < P_PAR; ++p)
            s += net_sum[(size_t)par[p] * B_BATCH + b];
    }
    const bool  pos = (s >= 0.f);
    const float ns  = s * (pos ? apos[n] : aneg[n]);
    const float o   = fmaf(s, (pos ? cpos[n] : cneg[n]), b2[n]);

    const size_t oi = (size_t)n * B_BATCH + b;
    net_sum[oi] = ns;
    outv[oi]    = o;
    fin[oi]     = o;     // seed max-pool with node's own output
}

// ---------------------------------------------------------------------------
// Kernel 13: segment max over child->parent edges (float atomicMax ->
// GLOBAL_ATOMIC_MAX_NUM_F32; max is order-independent -> deterministic).
// ---------------------------------------------------------------------------
__global__ __launch_bounds__(256) void segmax_kernel(const int* __restrict__ child_e,
                                                     const int* __restrict__ parent_e,
                                                     const float* __restrict__ outv,
                                                     float* __restrict__ fin)
{
    const int idx = blockIdx.x * blockDim.x + threadIdx.x;
    if (idx >= N_EDGE * B_BATCH) return;
    const int b = idx & (B_BATCH - 1);
    const int e = idx >> 6;
    const int c = child_e[e];
    const int p = parent_e[e];
    atomicMax(fin + (size_t)p * B_BATCH + b, outv[(size_t)c * B_BATCH + b]);
}

// ---------------------------------------------------------------------------
// Kernel 14: [N,B] -> [B,N] tiled transpose through LDS (coalesced both ways).
// ---------------------------------------------------------------------------
__global__ __launch_bounds__(256) void transpose_kernel(const float* __restrict__ fin,
                                                        float* __restrict__ dout)
{
    __shared__ float tile[64][65];
    const int n0  = blockIdx.x << 6;
    const int tid = threadIdx.x;

    #pragma unroll
    for (int i = 0; i < 16; ++i) {
        const int le = tid + (i << 8);
        const int nl = le >> 6;
        const int b  = le & 63;
        if (n0 + nl < N_NODES)
            tile[nl][b] = fin[(size_t)(n0 + nl) * B_BATCH + b];
    }
    __syncthreads();
    #pragma unroll
    for (int i = 0; i < 16; ++i) {
        const int le = tid + (i << 8);
        const int nl = le & 63;
        const int b  = le >> 6;
        if (n0 + nl < N_NODES)
            dout[(size_t)b * N_NODES + n0 + nl] = tile[nl][b];
    }
}

// ---------------------------------------------------------------------------
extern "C" void kernel_launch(void* const* d_in, const int* in_sizes, int n_in,
                              void* d_out, int out_size, void* d_ws, size_t ws_size,
                              hipStream_t stream)
{
    const float* inputs   = (const float*)d_in[0];   // [B, D]
    const float* w1       = (const float*)d_in[1];   // [N, H]
    const float* w2       = (const float*)d_in[2];   // [N, H]
    const float* b2       = (const float*)d_in[3];   // [N]
    const int*   parents  = (const int*)  d_in[4];   // [N, P]
    const int*   child_e  = (const int*)  d_in[5];   // [E]
    const int*   parent_e = (const int*)  d_in[6];   // [E]
    float* dout = (float*)d_out;                     // [B, N]

    float* ws      = (float*)d_ws;
    float* s_in    = ws;                                   // 64
    float* apos    = ws + 64;                              // N
    float* aneg    = apos + N_NODES;                       // N
    float* cpos    = aneg + N_NODES;                       // N
    float* cneg    = cpos + N_NODES;                       // N
    float* net_sum = cneg + N_NODES;                       // N*B
    float* outv    = net_sum + (size_t)N_NODES * B_BATCH;  // N*B
    float* fin     = outv    + (size_t)N_NODES * B_BATCH;  // N*B

    hipLaunchKernelGGL(rowsum_wmma_kernel, dim3(B_BATCH / 16), dim3(32), 0, stream,
                       inputs, s_in);
    hipLaunchKernelGGL(coeff_kernel, dim3(N_NODES / 8), dim3(256), 0, stream,
                       w1, w2, apos, aneg, cpos, cneg);
    for (int l = 0; l < L_LEV; ++l) {
        hipLaunchKernelGGL(level_kernel, dim3(S_LEV * B_BATCH / 256), dim3(256), 0, stream,
                           s_in, apos, aneg, cpos, cneg, b2, parents,
                           net_sum, outv, fin, l * S_LEV);
    }
    hipLaunchKernelGGL(segmax_kernel, dim3((N_EDGE * B_BATCH + 255) / 256), dim3(256), 0, stream,
                       child_e, parent_e, outv, fin);
    hipLaunchKernelGGL(transpose_kernel, dim3((N_NODES + 63) / 64), dim3(256), 0, stream,
                       fin, dout);
}